// GCN_18030272708828
// MI455X (gfx1250) — compile-verified
//
#include <hip/hip_runtime.h>
#include <stdint.h>

typedef __attribute__((ext_vector_type(2))) float v2f;
typedef __attribute__((ext_vector_type(8))) float v8f;

// ---------------------------------------------------------------------------
// Zero-fill (harness poisons d_out/d_ws with 0xAA; segment_sum needs zeros)
// ---------------------------------------------------------------------------
__global__ void zero_kernel(float* __restrict__ p, long long n) {
    long long i = (long long)blockIdx.x * blockDim.x + threadIdx.x;
    long long stride = (long long)gridDim.x * blockDim.x;
    for (; i < n; i += stride) p[i] = 0.0f;
}

// ---------------------------------------------------------------------------
// GEMM C[M,N] = A[M,K] * B[K,N] using V_WMMA_F32_16X16X4_F32 (wave32).
//
// Block = NT*MT waves (== 8 -> 256 threads).
// A rows are staged into LDS in K-chunks of KC with DOUBLE BUFFERING via the
// CDNA5 async global->LDS path (GLOBAL_LOAD_ASYNC_TO_LDS_B128, ASYNCcnt):
// chunk k+1 streams into buffer ^1 while WMMAs consume chunk k from buffer cur,
// then s_wait_asynccnt 0 + barrier. No VGPR round-trip for the A tile.
//
// LDS row stride = KC+4 floats: keeps 16B alignment for b128 async stores and
// gives bank stride 4 mod 64, so a fragment's 32 lanes hit 32 distinct banks.
//
// Fragment layouts per CDNA5 ISA 7.12.2:
//   A (16x4 f32):  lane L, VGPR v  ->  A[L%16][k0 + 2*(L>>4) + v]
//   B (4x16 f32):  lane L, VGPR v  ->  B[k0 + 2*(L>>4) + v][L%16]
//   C (16x16 f32): lane L, VGPR v  ->  C[v + 8*(L>>4)][L%16]
// ---------------------------------------------------------------------------
template <int K, int KC, int NT, int MT>
__launch_bounds__(256)
__global__ void gemm_wmma(const float* __restrict__ A,
                          const float* __restrict__ B,
                          float* __restrict__ C,
                          int M, int N) {
    constexpr int WAVES = NT * MT;          // == 8
    constexpr int LDA   = KC + 4;           // padded LDS row stride (floats)
    constexpr int BUF   = MT * 16 * LDA;    // floats per staging buffer
    static_assert(WAVES == 8, "block must be 8 waves");
    static_assert(K % KC == 0, "K must be a multiple of KC");
    static_assert(KC % 4 == 0, "KC must be a multiple of 4");

    __shared__ float As[2 * BUF];

    const int tid  = threadIdx.x;
    const int wave = tid >> 5;
    const int lane = tid & 31;
    const int half = lane >> 4;             // 0 or 1
    const int lrow = lane & 15;

    const int wm = wave / NT;               // wave's m-tile inside block
    const int wn = wave % NT;               // wave's n-tile inside block

    const int mBlockBase = blockIdx.x * (MT * 16);
    const int mBase = mBlockBase + wm * 16;
    const int nBase = (blockIdx.y * NT + wn) * 16;

    v8f acc = {0.f, 0.f, 0.f, 0.f, 0.f, 0.f, 0.f, 0.f};

    // 16 WMMAs (KC=64) / 8 WMMAs (KC=32) per chunk, accumulate-chained.
    auto compute_chunk = [&](const float* buf, int kc) {
#pragma unroll
        for (int k0 = 0; k0 < KC; k0 += 4) {
            // A fragment from LDS (8B-aligned pair -> ds_load_b64)
            v2f a = *(const v2f*)&buf[(wm * 16 + lrow) * LDA + k0 + 2 * half];
            // B fragment from global (W matrices are tiny & L2/L1-hot)
            const float* bp = B + (size_t)(kc + k0 + 2 * half) * N + nBase + lrow;
            v2f b;
            b.x = bp[0];
            b.y = bp[N];
            acc = __builtin_amdgcn_wmma_f32_16x16x4_f32(
                    false, a, false, b, (short)0, acc, false, false);
        }
    };

    const bool fullBlock = (mBlockBase + MT * 16) <= M;

    if (fullBlock) {
        // ---- async double-buffered pipeline (ASYNCcnt / TDM-class path) ----
        auto issue_chunk = [&](float* dstBuf, int kc) {
#pragma unroll
            for (int idx = tid; idx < MT * 16 * (KC / 4); idx += WAVES * 32) {
                const int r  = idx / (KC / 4);
                const int c4 = (idx % (KC / 4)) * 4;
                const float*   gp = A + (size_t)(mBlockBase + r) * K + kc + c4;
                // low 32 bits of the shared-aperture flat address == LDS byte
                // offset (ISA 10.2 aperture mapping)
                const uint32_t lo = (uint32_t)(uintptr_t)&dstBuf[r * LDA + c4];
                asm volatile("global_load_async_to_lds_b128 %0, %1, off"
                             :: "v"(lo), "v"(gp) : "memory");
            }
        };

        issue_chunk(As, 0);
        asm volatile("s_wait_asynccnt 0x0" ::: "memory");
        __syncthreads();

        int cur = 0;
        for (int kc = 0; kc < K; kc += KC) {
            if (kc + KC < K)
                issue_chunk(As + (cur ^ 1) * BUF, kc + KC);   // prefetch next
            compute_chunk(As + cur * BUF, kc);                // consume current
            asm volatile("s_wait_asynccnt 0x0" ::: "memory");
            __syncthreads();
            cur ^= 1;
        }
    } else {
        // ---- ragged tail block: synchronous zero-filled staging ------------
        for (int kc = 0; kc < K; kc += KC) {
            for (int idx = tid; idx < MT * 16 * (KC / 4); idx += WAVES * 32) {
                const int r  = idx / (KC / 4);
                const int c4 = (idx % (KC / 4)) * 4;
                const int gr = mBlockBase + r;
                float x0 = 0.f, x1 = 0.f, x2 = 0.f, x3 = 0.f;
                if (gr < M) {
                    const float4 v = *(const float4*)(A + (size_t)gr * K + kc + c4);
                    x0 = v.x; x1 = v.y; x2 = v.z; x3 = v.w;
                }
                float* s = &As[r * LDA + c4];
                s[0] = x0; s[1] = x1; s[2] = x2; s[3] = x3;
            }
            __syncthreads();
            if (mBase < M)                       // wave-uniform (EXEC all-1s)
                compute_chunk(As, kc);
            __syncthreads();
        }
    }

    if (mBase < M) {
#pragma unroll
        for (int v = 0; v < 8; ++v) {
            const int m = mBase + v + 8 * half;
            C[(size_t)m * N + nBase + lrow] = acc[v];
        }
    }
}

// ---------------------------------------------------------------------------
// Edge aggregation: out[dst[e]][f] += h[src[e]][f]
// One thread per (edge, 4 contiguous features): float4 gather + 4 atomics.
// Both h and out tables (<=102.4MB) are L2-resident, atomics resolve in L2.
// ---------------------------------------------------------------------------
template <int F>
__launch_bounds__(256)
__global__ void aggregate_kernel(const float* __restrict__ h,
                                 const int* __restrict__ src,
                                 const int* __restrict__ dst,
                                 float* __restrict__ out,
                                 int nE) {
    constexpr int TPE = F / 4;              // threads per edge
    const long long t = (long long)blockIdx.x * blockDim.x + threadIdx.x;
    const int e = (int)(t / TPE);
    if (e >= nE) return;
    const int f = (int)(t % TPE) * 4;

    const int s = src[e];
    const int d = dst[e];
    const float4 v = *(const float4*)(h + (size_t)s * F + f);
    float* o = out + (size_t)d * F + f;
    atomicAdd(o + 0, v.x);
    atomicAdd(o + 1, v.y);
    atomicAdd(o + 2, v.z);
    atomicAdd(o + 3, v.w);
}

// ---------------------------------------------------------------------------
// Orchestration
// inputs: features[100000,512] f32, edge_src[E] i32, edge_dst[E] i32,
//         W1[512,256], W2[256,256], W3[256,16]  (all f32)
// output: [100000,16] f32
// ---------------------------------------------------------------------------
extern "C" void kernel_launch(void* const* d_in, const int* in_sizes, int n_in,
                              void* d_out, int out_size, void* d_ws, size_t ws_size,
                              hipStream_t stream) {
    const float* feat = (const float*)d_in[0];
    const int*   src  = (const int*)d_in[1];
    const int*   dst  = (const int*)d_in[2];
    const float* W1   = (const float*)d_in[3];
    const float* W2   = (const float*)d_in[4];
    const float* W3   = (const float*)d_in[5];
    float* out = (float*)d_out;

    const int M = 100000;
    const int E = in_sizes[1];

    float* h   = (float*)d_ws;                 // [100000, 256]
    float* agg = h + (size_t)M * 256;          // [100000, 256]  (needs ~205MB ws)

    const int mt16 = M / 16;                   // 6250 (exact)

    // ---- Layer 1: h = feat @ W1 ; agg = scatter_sum(h[src] -> dst) --------
    gemm_wmma<512, 64, 8, 1><<<dim3(mt16, 2), 256, 0, stream>>>(feat, W1, h, M, 256);
    zero_kernel<<<2048, 256, 0, stream>>>(agg, (long long)M * 256);
    {
        long long threads = (long long)E * (256 / 4);
        int blocks = (int)((threads + 255) / 256);
        aggregate_kernel<256><<<blocks, 256, 0, stream>>>(h, src, dst, agg, E);
    }

    // ---- Layer 2: h = agg @ W2 ; agg = scatter_sum(h[src] -> dst) ---------
    gemm_wmma<256, 64, 8, 1><<<dim3(mt16, 2), 256, 0, stream>>>(agg, W2, h, M, 256);
    zero_kernel<<<2048, 256, 0, stream>>>(agg, (long long)M * 256);
    {
        long long threads = (long long)E * (256 / 4);
        int blocks = (int)((threads + 255) / 256);
        aggregate_kernel<256><<<blocks, 256, 0, stream>>>(h, src, dst, agg, E);
    }

    // ---- Layer 3: h3 = agg @ W3 ; out = scatter_sum(h3[src] -> dst) -------
    gemm_wmma<256, 32, 1, 8><<<dim3((mt16 + 7) / 8, 1), 256, 0, stream>>>(agg, W3, h, M, 16);
    zero_kernel<<<512, 256, 0, stream>>>(out, (long long)M * 16);
    {
        long long threads = (long long)E * (16 / 4);
        int blocks = (int)((threads + 255) / 256);
        aggregate_kernel<16><<<blocks, 256, 0, stream>>>(h, src, dst, out, E);
    }
}